// LinkPredictor_18648747999235
// MI455X (gfx1250) — compile-verified
//
#include <hip/hip_runtime.h>

typedef __attribute__((ext_vector_type(2))) float v2f;
typedef __attribute__((ext_vector_type(8))) float v8f;

#define IN_CH  128
#define HID_CH 256
#define OUT_CH 128

// ---------------- edge scatter: agg[dst] += feat[src], deg[dst] += 1 ----------------
__global__ __launch_bounds__(256)
void scatter_accum_kernel(const float* __restrict__ feat,
                          const int* __restrict__ src,
                          const int* __restrict__ dst,
                          float* __restrict__ agg,
                          float* __restrict__ deg,   // nullptr -> skip degree accumulation
                          int n_edges, int C)
{
    int gid  = blockIdx.x * blockDim.x + threadIdx.x;
    int e    = gid >> 5;
    int lane = gid & 31;
    if (e >= n_edges) return;
    int s = src[e], d = dst[e];
    const float* fs = feat + (size_t)s * C;
    float*       ad = agg  + (size_t)d * C;
    for (int c = lane; c < C; c += 32)
        atomicAdd(&ad[c], fs[c]);
    if (deg != nullptr && lane == 0)
        atomicAdd(&deg[d], 1.0f);
}

// ---------------- fused SAGE layer GEMM ----------------
// out[16-row tile] = act( (agg/deg) @ Wl + bl + xroot @ Wr )
// One block per 16-row tile of nodes; one wave32 per 16-column output tile.
// A/X tiles staged in LDS (padded stride -> conflict-free row reads).
template<int K, int NOUT, bool RELU>
__global__ __launch_bounds__((NOUT / 16) * 32)
void sage_gemm_kernel(const float* __restrict__ aggsum,  // [n,K] unscaled sums
                      const float* __restrict__ deg,     // [n]
                      const float* __restrict__ xroot,   // [n,K]
                      const float* __restrict__ Wl,      // [K,NOUT]
                      const float* __restrict__ bl,      // [NOUT]
                      const float* __restrict__ Wr,      // [K,NOUT]
                      float* __restrict__ out,           // [n,NOUT]
                      int n)
{
    constexpr int NT   = NOUT / 16;   // waves per block
    constexpr int LDSW = K + 4;       // padded row stride (float) -> banks spread
    __shared__ float sA[16 * LDSW];
    __shared__ float sX[16 * LDSW];

    const int m0  = blockIdx.x * 16;
    const int tid = threadIdx.x;

    // cooperative load of 16 x K tiles (mean-scale folded into A load)
    for (int i = tid; i < 16 * K; i += NT * 32) {
        int r = i / K, c = i - r * K;
        int row = m0 + r;
        float a = 0.0f, x = 0.0f;
        if (row < n) {
            float inv = 1.0f / fmaxf(deg[row], 1.0f);
            a = aggsum[(size_t)row * K + c] * inv;
            x = xroot [(size_t)row * K + c];
        }
        sA[r * LDSW + c] = a;
        sX[r * LDSW + c] = x;
    }
    __syncthreads();

    const int wave = tid >> 5;
    const int lane = tid & 31;
    const int rc   = lane & 15;          // A row (M) for SRC0; B col (N) for SRC1
    const int kh   = (lane >> 4) << 1;   // lanes 0-15: K+0/1, lanes 16-31: K+2/3
    const int n0   = wave * 16;

    v8f acc = {};
    const float* sArow = &sA[rc * LDSW];
    const float* sXrow = &sX[rc * LDSW];
    const float* wlcol = Wl + n0 + rc;
    const float* wrcol = Wr + n0 + rc;

    #pragma unroll 4
    for (int k = 0; k < K; k += 4) {     // (agg/deg) @ Wl
        v2f a, b;
        a.x = sArow[k + kh];
        a.y = sArow[k + kh + 1];
        b.x = wlcol[(size_t)(k + kh)     * NOUT];
        b.y = wlcol[(size_t)(k + kh + 1) * NOUT];
        acc = __builtin_amdgcn_wmma_f32_16x16x4_f32(false, a, false, b,
                                                    (short)0, acc, false, false);
    }
    #pragma unroll 4
    for (int k = 0; k < K; k += 4) {     // xroot @ Wr, accumulated into same C
        v2f a, b;
        a.x = sXrow[k + kh];
        a.y = sXrow[k + kh + 1];
        b.x = wrcol[(size_t)(k + kh)     * NOUT];
        b.y = wrcol[(size_t)(k + kh + 1) * NOUT];
        acc = __builtin_amdgcn_wmma_f32_16x16x4_f32(false, a, false, b,
                                                    (short)0, acc, false, false);
    }

    // C/D layout: VGPR r -> M = r (lanes 0-15) or r+8 (lanes 16-31), N = lane&15
    const float bias  = bl[n0 + rc];
    const int   rbase = (lane >> 4) << 3;
    #pragma unroll
    for (int r = 0; r < 8; r++) {
        int row = m0 + rbase + r;
        if (row < n) {
            float v = acc[r] + bias;
            if (RELU) v = fmaxf(v, 0.0f);
            out[(size_t)row * NOUT + n0 + rc] = v;
        }
    }
}

// ---------------- link prediction: out[e] = dot(z[lsrc], z[ldst]) ----------------
__global__ __launch_bounds__(256)
void edge_dot_kernel(const float* __restrict__ z,
                     const int* __restrict__ lsrc,
                     const int* __restrict__ ldst,
                     float* __restrict__ out, int n_label)
{
    int gid  = blockIdx.x * blockDim.x + threadIdx.x;
    int e    = gid >> 5;
    int lane = gid & 31;
    if (e >= n_label) return;
    const float4* a = (const float4*)(z + (size_t)lsrc[e] * OUT_CH);
    const float4* b = (const float4*)(z + (size_t)ldst[e] * OUT_CH);
    float4 av = a[lane], bv = b[lane];   // 32 lanes x float4 = 128 ch
    float s = av.x * bv.x + av.y * bv.y + av.z * bv.z + av.w * bv.w;
    #pragma unroll
    for (int off = 16; off > 0; off >>= 1)
        s += __shfl_down(s, off, 32);
    if (lane == 0) out[e] = s;
}

extern "C" void kernel_launch(void* const* d_in, const int* in_sizes, int n_in,
                              void* d_out, int out_size, void* d_ws, size_t ws_size,
                              hipStream_t stream)
{
    const float* x   = (const float*)d_in[0];
    const int*   ei  = (const int*)  d_in[1];
    const int*   eli = (const int*)  d_in[2];
    const float* Wl1 = (const float*)d_in[3];
    const float* bl1 = (const float*)d_in[4];
    const float* Wr1 = (const float*)d_in[5];
    const float* Wl2 = (const float*)d_in[6];
    const float* bl2 = (const float*)d_in[7];
    const float* Wr2 = (const float*)d_in[8];
    float* out = (float*)d_out;

    const int n_nodes = in_sizes[0] / IN_CH;
    const int n_edges = in_sizes[1] / 2;
    const int n_label = in_sizes[2] / 2;
    const int* src  = ei;
    const int* dst  = ei + n_edges;
    const int* lsrc = eli;
    const int* ldst = eli + n_label;

    // workspace layout (floats): deg | agg(N*256, reused both layers) | h(N*256) | z(N*128)
    float* deg = (float*)d_ws;
    float* agg = deg + n_nodes;
    float* h   = agg + (size_t)n_nodes * HID_CH;
    float* z   = h   + (size_t)n_nodes * HID_CH;

    const int mtiles = (n_nodes + 15) / 16;

    // ---- layer 1 ----
    hipMemsetAsync(deg, 0, (size_t)n_nodes * (1 + HID_CH) * sizeof(float), stream);
    {
        int blocks = (n_edges * 32 + 255) / 256;
        scatter_accum_kernel<<<blocks, 256, 0, stream>>>(x, src, dst, agg, deg,
                                                         n_edges, IN_CH);
    }
    sage_gemm_kernel<IN_CH, HID_CH, true>
        <<<mtiles, (HID_CH / 16) * 32, 0, stream>>>(agg, deg, x, Wl1, bl1, Wr1, h, n_nodes);

    // ---- layer 2 ----
    hipMemsetAsync(agg, 0, (size_t)n_nodes * HID_CH * sizeof(float), stream);
    {
        int blocks = (n_edges * 32 + 255) / 256;
        scatter_accum_kernel<<<blocks, 256, 0, stream>>>(h, src, dst, agg, nullptr,
                                                         n_edges, HID_CH);
    }
    sage_gemm_kernel<HID_CH, OUT_CH, false>
        <<<mtiles, (OUT_CH / 16) * 32, 0, stream>>>(agg, deg, h, Wl2, bl2, Wr2, z, n_nodes);

    // ---- link prediction dot ----
    {
        int blocks = (n_label * 32 + 255) / 256;
        edge_dot_kernel<<<blocks, 256, 0, stream>>>(z, lsrc, ldst, out, n_label);
    }
}